// MultiHeadAttention_19636590477892
// MI455X (gfx1250) — compile-verified
//
#include <hip/hip_runtime.h>
#include <math.h>

typedef __attribute__((ext_vector_type(2))) float v2f;
typedef __attribute__((ext_vector_type(8))) float v8f;

#define DMODEL 128
#define NHEAD  8
#define DHEAD  16

// ---------------------------------------------------------------------------
// Kernel 0: zero fill
// ---------------------------------------------------------------------------
__global__ __launch_bounds__(256) void zero_f32_kernel(float* __restrict__ p, size_t n) {
    size_t i = (size_t)blockIdx.x * 256 + threadIdx.x;
    size_t stride = (size_t)gridDim.x * 256;
    for (; i < n; i += stride) p[i] = 0.0f;
}

// ---------------------------------------------------------------------------
// Kernel 1: QKV projection via V_WMMA_F32_16X16X4_F32.
//   Q[n,j] = sum_k h[n,k] * W[j,k]   (x @ W.T)
// Block = 256 threads = 8 waves. blockIdx.x = 16-node tile, blockIdx.y = matrix.
// Each wave owns a 16x16 output tile (cols j0..j0+15), K-loop over 128 in 4s.
// A (16x4 f32): lane = (k>>1)*16 + m, vgpr = k&1  -> float2 per lane per step.
// B (4x16 f32): lane = (k>>1)*16 + n, vgpr = k&1  -> B[k][n] = W[j0+n][k].
// D (16x16 f32): vgpr r -> m = r + (lane>>4)*8, n = lane&15.
// ---------------------------------------------------------------------------
__global__ __launch_bounds__(256) void qkv_gemm_kernel(
    const float* __restrict__ hmat,
    const float* __restrict__ Wq,
    const float* __restrict__ Wk,
    const float* __restrict__ Wv,
    float* __restrict__ Qo,
    float* __restrict__ Ko,
    float* __restrict__ Vo)
{
    __shared__ float sA[16][DMODEL + 4];   // +4 pad: 132-float stride avoids bank conflicts

    const int mat = blockIdx.y;
    const float* W   = (mat == 0) ? Wq : (mat == 1) ? Wk : Wv;
    float*       Out = (mat == 0) ? Qo : (mat == 1) ? Ko : Vo;
    const int node0 = blockIdx.x * 16;

    // cooperative load of the 16x128 h tile (coalesced)
    for (int f = threadIdx.x; f < 16 * DMODEL; f += 256) {
        int r = f >> 7, c = f & (DMODEL - 1);
        sA[r][c] = hmat[(size_t)(node0 + r) * DMODEL + c];
    }
    __syncthreads();

    const int lane = threadIdx.x & 31;
    const int wave = threadIdx.x >> 5;
    const int j0   = wave * 16;          // this wave's output column tile
    const int ln16 = lane & 15;          // A: row m    B/D: col n
    const int grp  = lane >> 4;          // K-group select (0 -> k,k+1 ; 1 -> k+2,k+3)

    const float* wrow = W + (size_t)(j0 + ln16) * DMODEL;  // row j of W (B column n)

    v8f acc = {};
    #pragma unroll
    for (int k0 = 0; k0 < DMODEL; k0 += 4) {
        const int kb = k0 + (grp << 1);
        v2f a = *(const v2f*)&sA[ln16][kb];
        v2f b = *(const v2f*)&wrow[kb];
        acc = __builtin_amdgcn_wmma_f32_16x16x4_f32(
            /*neg_a=*/false, a, /*neg_b=*/false, b,
            /*c_mod=*/(short)0, acc, /*reuse_a=*/false, /*reuse_b=*/false);
    }

    #pragma unroll
    for (int r = 0; r < 8; ++r) {
        const int m = r + (grp << 3);
        Out[(size_t)(node0 + m) * DMODEL + j0 + ln16] = acc[r];
    }
}

// ---------------------------------------------------------------------------
// Kernel 2: edge scores  s = exp(clip(dot(K[src],Q[dst])/sqrt(16), -5, 5))
// plus segment-max into m[dst,h] (uint atomicMax valid: s > 0).
// One thread per (edge, head).
// ---------------------------------------------------------------------------
__global__ __launch_bounds__(256) void edge_score_kernel(
    const float* __restrict__ Q, const float* __restrict__ K,
    const int* __restrict__ src, const int* __restrict__ dst,
    float* __restrict__ s_out, unsigned int* __restrict__ m_out, int EH)
{
    int idx = blockIdx.x * 256 + threadIdx.x;
    if (idx >= EH) return;
    const int e = idx >> 3, h = idx & 7;
    const int sn = src[e], dn = dst[e];

    const float4* kp = (const float4*)(K + (size_t)sn * DMODEL + h * DHEAD);
    const float4* qp = (const float4*)(Q + (size_t)dn * DMODEL + h * DHEAD);
    float acc = 0.0f;
    #pragma unroll
    for (int i = 0; i < 4; ++i) {
        float4 kk = kp[i], qq = qp[i];
        acc += kk.x * qq.x + kk.y * qq.y + kk.z * qq.z + kk.w * qq.w;
    }
    acc *= 0.25f;                              // 1/sqrt(DH), DH=16
    acc = fminf(5.0f, fmaxf(-5.0f, acc));
    const float s = expf(acc);
    s_out[idx] = s;
    atomicMax(&m_out[(size_t)dn * NHEAD + h], __float_as_uint(s));
}

// ---------------------------------------------------------------------------
// Kernel 3: z[dst,h] = segment_sum(exp(s - m[dst,h]))
// ---------------------------------------------------------------------------
__global__ __launch_bounds__(256) void edge_z_kernel(
    const int* __restrict__ dst,
    const float* __restrict__ s_in, const float* __restrict__ m_in,
    float* __restrict__ z_out, int EH)
{
    int idx = blockIdx.x * 256 + threadIdx.x;
    if (idx >= EH) return;
    const int e = idx >> 3, h = idx & 7;
    const int dn = dst[e];
    const float m  = m_in[(size_t)dn * NHEAD + h];
    const float ee = expf(s_in[idx] - m);
    atomicAdd(&z_out[(size_t)dn * NHEAD + h], ee);
}

// ---------------------------------------------------------------------------
// Kernel 4: out[dst] += alpha * V[src]   (alpha recomputed -> no extra buffer)
// ---------------------------------------------------------------------------
__global__ __launch_bounds__(256) void edge_agg_kernel(
    const int* __restrict__ src, const int* __restrict__ dst,
    const float* __restrict__ V,
    const float* __restrict__ s_in, const float* __restrict__ m_in,
    const float* __restrict__ z_in,
    float* __restrict__ out, int EH)
{
    int idx = blockIdx.x * 256 + threadIdx.x;
    if (idx >= EH) return;
    const int e = idx >> 3, h = idx & 7;
    const int sn = src[e], dn = dst[e];
    const float m = m_in[(size_t)dn * NHEAD + h];
    const float z = z_in[(size_t)dn * NHEAD + h];
    const float alpha = expf(s_in[idx] - m) / z;

    const float4* vp = (const float4*)(V + (size_t)sn * DMODEL + h * DHEAD);
    float* op = out + (size_t)dn * DMODEL + h * DHEAD;
    #pragma unroll
    for (int i = 0; i < 4; ++i) {
        float4 v = vp[i];
        atomicAdd(op + i * 4 + 0, alpha * v.x);
        atomicAdd(op + i * 4 + 1, alpha * v.y);
        atomicAdd(op + i * 4 + 2, alpha * v.z);
        atomicAdd(op + i * 4 + 3, alpha * v.w);
    }
}

// ---------------------------------------------------------------------------
// Kernel 5: nodes with no in-edges (z == 0) keep their V projection
// ---------------------------------------------------------------------------
__global__ __launch_bounds__(256) void finalize_kernel(
    const float* __restrict__ V, const float* __restrict__ z_in,
    float* __restrict__ out, int NH)
{
    int idx = blockIdx.x * 256 + threadIdx.x;
    if (idx >= NH) return;
    if (z_in[idx] == 0.0f) {
        const int n = idx >> 3, h = idx & 7;
        const float4* vp = (const float4*)(V + (size_t)n * DMODEL + h * DHEAD);
        float4* op = (float4*)(out + (size_t)n * DMODEL + h * DHEAD);
        #pragma unroll
        for (int i = 0; i < 4; ++i) op[i] = vp[i];
    }
}

// ---------------------------------------------------------------------------
extern "C" void kernel_launch(void* const* d_in, const int* in_sizes, int n_in,
                              void* d_out, int out_size, void* d_ws, size_t ws_size,
                              hipStream_t stream) {
    const float* h_in = (const float*)d_in[0];
    const float* Wq   = (const float*)d_in[1];
    const float* Wk   = (const float*)d_in[2];
    const float* Wv   = (const float*)d_in[3];
    const int*   src  = (const int*)d_in[4];
    const int*   dst  = (const int*)d_in[5];
    float* out = (float*)d_out;

    const int N  = in_sizes[0] / DMODEL;   // 50000
    const int E  = in_sizes[4];            // 1600000
    const int EH = E * NHEAD;
    const int NH = N * NHEAD;

    // workspace layout (floats)
    float* Q = (float*)d_ws;
    float* K = Q + (size_t)N * DMODEL;
    float* V = K + (size_t)N * DMODEL;
    float* s = V + (size_t)N * DMODEL;
    float* m = s + (size_t)EH;
    float* z = m + (size_t)NH;

    // 0) zero out / m / z  (m,z contiguous)
    zero_f32_kernel<<<1024, 256, 0, stream>>>(out, (size_t)N * DMODEL);
    zero_f32_kernel<<<256, 256, 0, stream>>>(m, (size_t)2 * NH);

    // 1) QKV projections (WMMA). N is a multiple of 16 (50000 = 3125*16).
    dim3 ggrid(N / 16, 3);
    qkv_gemm_kernel<<<ggrid, 256, 0, stream>>>(h_in, Wq, Wk, Wv, Q, K, V);

    const int eblocks = (EH + 255) / 256;
    // 2) edge scores + segment max
    edge_score_kernel<<<eblocks, 256, 0, stream>>>(Q, K, src, dst, s,
                                                   (unsigned int*)m, EH);
    // 3) segment sum of exp(s - m)
    edge_z_kernel<<<eblocks, 256, 0, stream>>>(dst, s, m, z, EH);
    // 4) weighted aggregation
    edge_agg_kernel<<<eblocks, 256, 0, stream>>>(src, dst, V, s, m, z, out, EH);
    // 5) no-in-edge nodes fall back to V
    finalize_kernel<<<(NH + 255) / 256, 256, 0, stream>>>(V, z, out, NH);
}